// RandomShiftsAug_40183714021701
// MI455X (gfx1250) — compile-verified
//
#include <hip/hip_runtime.h>
#include <cstdint>

// RandomShiftsAug for x:[512,9,84,84] f32, shift_int:[512,1,1,2] i32 in [0,8].
// Grid math collapses to an integer clamped shift:
//   out[n,c,i,j] = x[n,c, clamp(i+sy-4,0,83), clamp(j+sx-4,0,83)]
// Pure HBM-bandwidth-bound (~260 MB total => ~11us floor @ 23.3 TB/s).
// CDNA5 strategy: one Tensor Data Mover DMA per workgroup stages the whole
// 28.2 KB channel image global->LDS (TENSORcnt path), waves gather the
// shifted/clamped pixels out of LDS, store aligned float4 non-temporally.

#define NB    512
#define CC    9
#define HW    84
#define IMG   (HW * HW)     // 7056 floats per channel image
#define VEC4  (IMG / 4)     // 1764 float4 per channel image
#define PADV  4
#define TPB   256

typedef float        v4f __attribute__((ext_vector_type(4)));
typedef unsigned int v4u __attribute__((ext_vector_type(4)));
typedef unsigned int v8u __attribute__((ext_vector_type(8)));

__device__ __forceinline__ int clampi(int v, int lo, int hi) {
    v = v < lo ? lo : v;
    return v > hi ? hi : v;
}

__global__ __launch_bounds__(TPB) void RandomShiftsAug_kernel(
    const float* __restrict__ x,
    const int*   __restrict__ shift_int,
    float*       __restrict__ out)
{
    __shared__ float smem[IMG];   // 28224 B: one channel image

    const int nc  = blockIdx.x;        // 0 .. 512*9-1
    const int n   = nc / CC;
    const int tid = threadIdx.x;

    // per-batch integer shift; channel 0 = x(width), channel 1 = y(height)
    const int dx = shift_int[2 * n + 0] - PADV;   // [-4, 4]
    const int dy = shift_int[2 * n + 1] - PADV;   // [-4, 4]

    // ---- Stage: one TDM DMA of the whole tile, issued by wave 0 only ----
    if (tid < 32) {
        unsigned long long ga =
            (unsigned long long)(uintptr_t)(x + (size_t)nc * IMG);
        unsigned ga_lo    = __builtin_amdgcn_readfirstlane((unsigned)(ga & 0xFFFFFFFFull));
        unsigned ga_hi    = __builtin_amdgcn_readfirstlane((unsigned)(ga >> 32));
        unsigned lds_base = __builtin_amdgcn_readfirstlane((unsigned)(uintptr_t)&smem[0]);

        // D# group 0 (128b): count=1 (valid), lds_addr, global_addr[56:0], type=2
        v4u g0;
        g0.x = 1u;                                  // count=1, is_restore=0, gather=0
        g0.y = lds_base;                            // lds_addr (bytes)
        g0.z = ga_lo;                               // global_addr[31:0]
        g0.w = (ga_hi & 0x01FFFFFFu) | (2u << 30);  // global_addr[56:32] | type=2

        // D# group 1 (256b): 1-D tile covering the whole channel image
        v8u g1;
        g1.s0 = 2u << 16;                           // wg_mask=0, data_size=2 (4 bytes)
        g1.s1 = ((unsigned)IMG & 0xFFFFu) << 16;    // tensor_dim0[15:0]  @ bits[63:48]
        g1.s2 = ((unsigned)IMG >> 16) | (1u << 16); // tensor_dim0[31:16] | tensor_dim1=1
        g1.s3 = ((unsigned)IMG & 0xFFFFu) << 16;    // tile_dim0 = 7056   @ bits[127:112]
        g1.s4 = 1u;                                 // tile_dim1=1, tile_dim2 unused
        g1.s5 = (unsigned)IMG;                      // tensor_dim0_stride[31:0]
        g1.s6 = 0u;                                 // stride hi / tensor_dim1_stride lo
        g1.s7 = 0u;                                 // tensor_dim1_stride hi

        // 2-operand form: groups 2/3 disabled (tensor <= 2D)
        asm volatile("tensor_load_to_lds %0, %1" :: "s"(g0), "s"(g1) : "memory");
        asm volatile("s_wait_tensorcnt 0x0" ::: "memory");
    }
    __syncthreads();   // publish LDS tile to all 8 waves

    // ---- Gather: shifted + clamped reads from LDS, b128 NT stores ----
    float* dst = out + (size_t)nc * IMG;
    for (int t = tid; t < VEC4; t += TPB) {
        const int flat = 4 * t;
        const int i = flat / HW;            // output row (84 % 4 == 0: a vec4
        const int j = flat - i * HW;        // output col  never spans rows)

        const int iy = clampi(i + dy, 0, HW - 1);
        const float* row = &smem[iy * HW];

        v4f v;
        v.x = row[clampi(j + 0 + dx, 0, HW - 1)];
        v.y = row[clampi(j + 1 + dx, 0, HW - 1)];
        v.z = row[clampi(j + 2 + dx, 0, HW - 1)];
        v.w = row[clampi(j + 3 + dx, 0, HW - 1)];

        __builtin_nontemporal_store(v, (v4f*)(dst + flat));
    }
}

extern "C" void kernel_launch(void* const* d_in, const int* in_sizes, int n_in,
                              void* d_out, int out_size, void* d_ws, size_t ws_size,
                              hipStream_t stream)
{
    const float* x         = (const float*)d_in[0];   // [512,9,84,84] f32
    const int*   shift_int = (const int*)d_in[1];     // [512,1,1,2] i32
    float*       out       = (float*)d_out;           // [512,9,84,84] f32

    (void)in_sizes; (void)n_in; (void)out_size; (void)d_ws; (void)ws_size;

    dim3 grid(NB * CC);   // one workgroup per (n, c) channel image
    dim3 block(TPB);      // 8 wave32 waves
    RandomShiftsAug_kernel<<<grid, block, 0, stream>>>(x, shift_int, out);
}